// Attention_60662118088811
// MI455X (gfx1250) — compile-verified
//
#include <hip/hip_runtime.h>
#include <hip/hip_bf16.h>
#include <math.h>

// Problem constants (match reference)
#define B 32
#define T 2048
#define D 512

typedef __attribute__((ext_vector_type(2))) float v2f;
typedef __attribute__((ext_vector_type(8))) float v8f;
typedef unsigned int u32x4 __attribute__((ext_vector_type(4)));
typedef unsigned int u32x8 __attribute__((ext_vector_type(8)));

// Energy kernel tiling: each 256-thread WG owns 128 t-rows of one batch.
// D is processed in 4 chunks of 128 columns staged through LDS by the TDM.
#define ROWS    128          // t-rows per workgroup
#define DC      128          // D-columns per chunk
#define LSTRIDE (DC + 4)     // padded LDS row stride (floats) -> bank-conflict free

// -----------------------------------------------------------------------------
// Kernel 1: energy[b,t] = dot(key[b,t,:], query[b,:])
//
// TDM (tensor_load_to_lds) DMAs a 128x128 fp32 tile of key into LDS with a
// 4-dword pad after each 128-dword row (pad_interval=128dw, pad_amount=4dw),
// giving row stride 132 floats. WMMA A-operand reads then hit distinct banks:
// lane L reads row (L&15) at col 4k+2*(L>>4) -> bank 4*row + col, all unique.
//
// Math per wave (owns 16 rows): V_WMMA_F32_16X16X4_F32 with the query chunk
// broadcast into all 16 B-columns (every output column equals the dot product;
// redundant compute, identical bandwidth, full fp32 accumulation).
// -----------------------------------------------------------------------------
__global__ __launch_bounds__(256) void energy_kernel(
    const float* __restrict__ q,   // [B, D]
    const float* __restrict__ k,   // [B, T, D]
    float* __restrict__ energy)    // [B, T]
{
    __shared__ float tile[ROWS * LSTRIDE];   // 67,584 bytes

    const int lane = threadIdx.x & 31;
    const int wave = threadIdx.x >> 5;       // 8 waves, wave w owns rows 16w..16w+15
    const int b    = blockIdx.x >> 4;        // 16 row-tiles per batch
    const int t0   = (blockIdx.x & 15) * ROWS;

    const float* kbase = k + ((size_t)b * T + t0) * D;   // tile origin in key
    const float* qrow  = q + (size_t)b * D;

    const int colOff = (lane >> 4) << 1;                     // 0 | 2
    const int rbase  = (wave * 16 + (lane & 15)) * LSTRIDE;  // LDS row for this lane

    v8f acc = {0.f, 0.f, 0.f, 0.f, 0.f, 0.f, 0.f, 0.f};

    // LDS byte offset of the tile (flat shared address: low 32 bits = LDS offset)
    const unsigned ldsoff = (unsigned)(uintptr_t)(&tile[0]);

    for (int c = 0; c < D / DC; ++c) {
        __syncthreads();   // previous chunk fully consumed before overwrite

        if (wave == 0) {
            // ---- Tensor DMA Descriptor (D#), 2D tensor -> groups 0 and 1 ----
            const unsigned long long ga =
                (unsigned long long)(uintptr_t)(kbase + c * DC);

            u32x4 g0;
            g0[0] = 1u;                                  // count=1, user mode
            g0[1] = ldsoff;                              // lds_addr (bytes)
            g0[2] = (unsigned)(ga & 0xFFFFFFFFu);        // global_addr[31:0]
            g0[3] = (unsigned)((ga >> 32) & 0x01FFFFFFu) // global_addr[56:32]
                  | (2u << 30);                          // type = 2 ("image")

            u32x8 g1;
            g1[0] = (2u << 16)      // data_size = 4 bytes
                  | (1u << 20)      // pad_enable
                  | (6u << 22)      // pad_interval: 128 dwords
                  | (3u << 25);     // pad_amount:   4 dwords
            g1[1] = ((unsigned)D & 0xFFFFu) << 16;       // tensor_dim0 lo16 (=512)
            g1[2] = ((unsigned)D >> 16)                  // tensor_dim0 hi16 (=0)
                  | (((unsigned)T & 0xFFFFu) << 16);     // tensor_dim1 lo16 (=2048)
            g1[3] = ((unsigned)T >> 16)                  // tensor_dim1 hi16 (=0)
                  | ((unsigned)DC << 16);                // tile_dim0 = 128 cols
            g1[4] = (unsigned)ROWS;                      // tile_dim1 = 128 rows; tile_dim2=0
            g1[5] = (unsigned)D;                         // tensor_dim0_stride lo32 (=512)
            g1[6] = 0u;                                  // stride0 hi16 | stride1 lo16
            g1[7] = 0u;                                  // stride1 hi32

            // 2D tensor: VADDR2/VADDR3 are NULL -> two-group form.
            asm volatile("tensor_load_to_lds %0, %1"
                         :
                         : "s"(g0), "s"(g1)
                         : "memory");
            __builtin_amdgcn_s_wait_tensorcnt(0);
        }
        __syncthreads();   // tile visible to all waves

        const float* qc = qrow + c * DC;
#pragma unroll 8
        for (int kk = 0; kk < DC / 4; ++kk) {
            const int cc = (kk << 2) + colOff;
            v2f a  = *(const v2f*)(&tile[rbase + cc]);   // ds_load_b64, conflict-free
            v2f bb = *(const v2f*)(qc + cc);             // query, cache-resident
            acc = __builtin_amdgcn_wmma_f32_16x16x4_f32(
                /*neg_a=*/false, a, /*neg_b=*/false, bb,
                /*c_mod=*/(short)0, acc, /*reuse_a=*/false, /*reuse_b=*/false);
        }
    }

    // Every column of the 16x16 result is identical; column 0 of each lane half
    // writes its 8 rows. Lane 0 -> rows +0..7, lane 16 -> rows +8..15.
    if ((lane & 15) == 0) {
        float* e = energy + (size_t)b * T + t0 + wave * 16 + ((lane >> 4) << 3);
#pragma unroll
        for (int i = 0; i < 8; ++i) e[i] = acc[i];
    }
}

// -----------------------------------------------------------------------------
// Kernel 2: attention[b,:] = softmax(energy[b,:]) over T. One 256-thread block
// per batch row; 8 elements/thread; LDS tree reductions for max and sum.
// -----------------------------------------------------------------------------
__global__ __launch_bounds__(256) void softmax_kernel(
    const float* __restrict__ energy,  // [B, T]
    float* __restrict__ attn)          // [B, T]
{
    const int b   = blockIdx.x;
    const int tid = threadIdx.x;
    __shared__ float red[256];

    float e[8];
    float m = -INFINITY;
#pragma unroll
    for (int i = 0; i < 8; ++i) {
        e[i] = energy[(size_t)b * T + tid + i * 256];
        m = fmaxf(m, e[i]);
    }
    red[tid] = m;
    __syncthreads();
    for (int s = 128; s > 0; s >>= 1) {
        if (tid < s) red[tid] = fmaxf(red[tid], red[tid + s]);
        __syncthreads();
    }
    m = red[0];
    __syncthreads();

    float sum = 0.f;
#pragma unroll
    for (int i = 0; i < 8; ++i) {
        e[i] = expf(e[i] - m);
        sum += e[i];
    }
    red[tid] = sum;
    __syncthreads();
    for (int s = 128; s > 0; s >>= 1) {
        if (tid < s) red[tid] += red[tid + s];
        __syncthreads();
    }
    const float inv = 1.0f / red[0];

#pragma unroll
    for (int i = 0; i < 8; ++i)
        attn[(size_t)b * T + tid + i * 256] = e[i] * inv;
}

// -----------------------------------------------------------------------------
// Kernel 3a: zero the context output region (d_out is poisoned by the harness).
// -----------------------------------------------------------------------------
__global__ void zero_kernel(float* __restrict__ p, int n)
{
    int i = blockIdx.x * blockDim.x + threadIdx.x;
    if (i < n) p[i] = 0.f;
}

// -----------------------------------------------------------------------------
// Kernel 3b: out[b,:] = sum_t attn[b,t] * value[b,t,:].
// 16-way split over T for occupancy (512 WGs). Each thread owns a float2
// column pair -> a wave reads a contiguous 256B slab per row (coalesced).
// global_prefetch_b8 streams 16 rows ahead. Partials combined with
// global_atomic_add_f32.
// -----------------------------------------------------------------------------
#define SPLIT 16
__global__ __launch_bounds__(256) void context_kernel(
    const float* __restrict__ attn,   // [B, T]
    const float* __restrict__ value,  // [B, T, D]
    float* __restrict__ out)          // [B, D]
{
    const int b   = blockIdx.x / SPLIT;
    const int seg = blockIdx.x % SPLIT;
    const int tid = threadIdx.x;       // 256 threads * 2 cols = D
    const int c   = tid * 2;
    const int rows = T / SPLIT;        // 128

    const float* abase = attn + (size_t)b * T + (size_t)seg * rows;
    const float* vbase = value + ((size_t)b * T + (size_t)seg * rows) * D;

    float ax = 0.f, ay = 0.f;
    for (int t = 0; t < rows; ++t) {
        __builtin_prefetch(vbase + (size_t)(t + 16) * D + c, 0, 1);
        const float  a = abase[t];
        const float2 v = *(const float2*)(vbase + (size_t)t * D + c);
        ax = fmaf(a, v.x, ax);
        ay = fmaf(a, v.y, ay);
    }
    float* o = out + (size_t)b * D + c;
    unsafeAtomicAdd(o + 0, ax);
    unsafeAtomicAdd(o + 1, ay);
}

// -----------------------------------------------------------------------------
// Launch: d_in = {query [B,D], key [B,T,D], value [B,T,D]} (fp32)
// d_out = out [B,D] ++ attention [B,T] (fp32).  d_ws: energy [B,T] scratch.
// -----------------------------------------------------------------------------
extern "C" void kernel_launch(void* const* d_in, const int* in_sizes, int n_in,
                              void* d_out, int out_size, void* d_ws, size_t ws_size,
                              hipStream_t stream)
{
    const float* q = (const float*)d_in[0];
    const float* k = (const float*)d_in[1];
    const float* v = (const float*)d_in[2];

    float* out    = (float*)d_out;       // [B, D]
    float* attn   = out + B * D;         // [B, T]
    float* energy = (float*)d_ws;        // [B, T] scratch (256 KB)

    // 1) energy = K . q  (TDM-staged LDS tiles + fp32 WMMA; reads 128 MB)
    //    B*T/ROWS = 512 workgroups of 8 waves.
    energy_kernel<<<(B * T) / ROWS, 256, 0, stream>>>(q, k, energy);

    // 3a) zero the out region before atomic accumulation
    zero_kernel<<<(B * D + 255) / 256, 256, 0, stream>>>(out, B * D);

    // 2) attention = softmax(energy) along T
    softmax_kernel<<<B, 256, 0, stream>>>(energy, attn);

    // 3b) out = attention . V  (reads 128 MB of value)
    context_kernel<<<B * SPLIT, 256, 0, stream>>>(attn, v, out);
}